// OFPenalty_6433861009619
// MI455X (gfx1250) — compile-verified
//
#include <hip/hip_runtime.h>

typedef __attribute__((ext_vector_type(2))) float v2f;
typedef __attribute__((ext_vector_type(8))) float v8f;

#define B_DIM 64
#define C_DIM 512
#define HW    784
#define EPS   1e-12f

// ---------------------------------------------------------------------------
// Kernel 1: batched AAT[b] = Wm[b] * Wm[b]^T via V_WMMA_F32_16X16X4_F32
// Each wave computes one 16-row M-strip against 4 N-tiles (16x64 of output),
// reusing the A fragment across the 4 WMMAs per K-step of 4.
// A-frag layout (16x4 f32): lane holds row = lane&15, K = 2*(lane>>4)+{0,1}
// B-frag layout (4x16 f32): lane holds col = lane&15, K = 2*(lane>>4)+{0,1}
//   -> for Wm*Wm^T both fragments load Wm rows with the identical pattern.
// D layout (16x16 f32): lane holds col = lane&15, vgpr v -> row v + 8*(lane>>4)
// ---------------------------------------------------------------------------
__global__ __launch_bounds__(128)
void of_aat_wmma(const float* __restrict__ A, float* __restrict__ AAT) {
    const int lane  = threadIdx.x & 31;
    const int wave  = threadIdx.x >> 5;       // 0..3
    const int m     = lane & 15;
    const int khalf = lane >> 4;              // 0 or 1

    const int b  = blockIdx.z;                // 0..63
    const int mt = blockIdx.x * 4 + wave;     // 0..31  (M tile)
    const int ng = blockIdx.y;                // 0..7   (group of 4 N tiles)

    const float* __restrict__ Wm = A + (size_t)b * C_DIM * HW;

    const float* __restrict__ ap  = Wm + (size_t)(mt * 16 + m) * HW + 2 * khalf;
    const float* __restrict__ bp0 = Wm + (size_t)((ng * 4 + 0) * 16 + m) * HW + 2 * khalf;
    const float* __restrict__ bp1 = Wm + (size_t)((ng * 4 + 1) * 16 + m) * HW + 2 * khalf;
    const float* __restrict__ bp2 = Wm + (size_t)((ng * 4 + 2) * 16 + m) * HW + 2 * khalf;
    const float* __restrict__ bp3 = Wm + (size_t)((ng * 4 + 3) * 16 + m) * HW + 2 * khalf;

    v8f acc0 = {}; v8f acc1 = {}; v8f acc2 = {}; v8f acc3 = {};

    #pragma unroll 4
    for (int k = 0; k < HW; k += 4) {
        v2f a  = *(const v2f*)(ap  + k);
        v2f b0 = *(const v2f*)(bp0 + k);
        v2f b1 = *(const v2f*)(bp1 + k);
        v2f b2 = *(const v2f*)(bp2 + k);
        v2f b3 = *(const v2f*)(bp3 + k);
        acc0 = __builtin_amdgcn_wmma_f32_16x16x4_f32(false, a, false, b0, (short)0, acc0, false, false);
        acc1 = __builtin_amdgcn_wmma_f32_16x16x4_f32(false, a, false, b1, (short)0, acc1, false, false);
        acc2 = __builtin_amdgcn_wmma_f32_16x16x4_f32(false, a, false, b2, (short)0, acc2, false, false);
        acc3 = __builtin_amdgcn_wmma_f32_16x16x4_f32(false, a, false, b3, (short)0, acc3, false, false);
    }

    // D: row = mt*16 + v + 8*khalf ; col = nt*16 + (lane&15)
    float* __restrict__ Cbase = AAT + (size_t)b * C_DIM * C_DIM
                              + (size_t)(mt * 16 + 8 * khalf) * C_DIM + m;
    {
        float* cp = Cbase + (size_t)(ng * 4 + 0) * 16;
        #pragma unroll
        for (int v = 0; v < 8; ++v) cp[(size_t)v * C_DIM] = acc0[v];
    }
    {
        float* cp = Cbase + (size_t)(ng * 4 + 1) * 16;
        #pragma unroll
        for (int v = 0; v < 8; ++v) cp[(size_t)v * C_DIM] = acc1[v];
    }
    {
        float* cp = Cbase + (size_t)(ng * 4 + 2) * 16;
        #pragma unroll
        for (int v = 0; v < 8; ++v) cp[(size_t)v * C_DIM] = acc2[v];
    }
    {
        float* cp = Cbase + (size_t)(ng * 4 + 3) * 16;
        #pragma unroll
        for (int v = 0; v < 8; ++v) cp[(size_t)v * C_DIM] = acc3[v];
    }
}

// ---------------------------------------------------------------------------
// Kernel 2: per-batch power iteration + Rayleigh quotients (AAT is L2-resident)
// ---------------------------------------------------------------------------
__device__ __forceinline__ float block_reduce_sum(float v, float* red) {
    const int t = threadIdx.x;
    red[t] = v;
    __syncthreads();
    #pragma unroll
    for (int s = 128; s > 0; s >>= 1) {
        if (t < s) red[t] += red[t + s];
        __syncthreads();
    }
    float r = red[0];
    __syncthreads();
    return r;
}

__device__ __forceinline__ void matvec512(const float* __restrict__ M,
                                          const float* __restrict__ xs,
                                          float* __restrict__ ys) {
    const int lane = threadIdx.x & 31;
    const int wave = threadIdx.x >> 5;       // 0..7, each wave does 64 rows
    for (int r = wave * 64; r < wave * 64 + 64; ++r) {
        const float* __restrict__ row = M + (size_t)r * C_DIM;
        float s = 0.0f;
        #pragma unroll 4
        for (int k = lane; k < C_DIM; k += 32) s += row[k] * xs[k];
        #pragma unroll
        for (int off = 16; off > 0; off >>= 1) s += __shfl_down(s, off, 32);
        if (lane == 0) ys[r] = s;
    }
    __syncthreads();
}

__global__ __launch_bounds__(256)
void of_power_iter(const float* __restrict__ AAT, const float* __restrict__ x0,
                   float* __restrict__ pen) {
    __shared__ float xs[C_DIM];
    __shared__ float ys[C_DIM];
    __shared__ float red[256];

    const int t = threadIdx.x;
    const int b = blockIdx.x;
    const float* __restrict__ M = AAT + (size_t)b * C_DIM * C_DIM;

    // x = normalize(x0)
    float p = 0.0f;
    for (int i = t; i < C_DIM; i += 256) {
        float v = x0[(size_t)b * C_DIM + i];
        xs[i] = v;
        p += v * v;
    }
    float n2 = block_reduce_sum(p, red);
    float inv = 1.0f / fmaxf(sqrtf(n2), EPS);
    for (int i = t; i < C_DIM; i += 256) xs[i] *= inv;
    __syncthreads();

    // 4 power-iteration steps: x = normalize(AAT @ x)   (3 no-grad + 1)
    for (int it = 0; it < 4; ++it) {
        matvec512(M, xs, ys);
        float q = 0.0f;
        for (int i = t; i < C_DIM; i += 256) { float v = ys[i]; q += v * v; }
        float n2y = block_reduce_sum(q, red);
        float iy = 1.0f / fmaxf(sqrtf(n2y), EPS);
        for (int i = t; i < C_DIM; i += 256) xs[i] = ys[i] * iy;
        __syncthreads();
    }
    // xs = x1
    matvec512(M, xs, ys);                     // ys = AAT @ x1
    float pn = 0.0f, pd = 0.0f;
    for (int i = t; i < C_DIM; i += 256) { pn += ys[i] * xs[i]; pd += xs[i] * xs[i]; }
    float num = block_reduce_sum(pn, red);
    float den = block_reduce_sum(pd, red);
    float largest = num / den;

    // x2 = normalize(A2 @ x1) = normalize(AAT@x1 - largest*x1)   (reuse ys)
    float q = 0.0f;
    for (int i = t; i < C_DIM; i += 256) {
        float v = ys[i] - largest * xs[i];
        ys[i] = v;
        q += v * v;
    }
    float n2z = block_reduce_sum(q, red);
    float iz = 1.0f / fmaxf(sqrtf(n2z), EPS);
    for (int i = t; i < C_DIM; i += 256) xs[i] = ys[i] * iz;
    __syncthreads();

    // tmp = x2^T A2 x2 / (x2.x2) = (x2.(AAT x2) - largest*(x2.x2)) / (x2.x2)
    matvec512(M, xs, ys);                     // ys = AAT @ x2
    float pn2 = 0.0f, pd2 = 0.0f;
    for (int i = t; i < C_DIM; i += 256) { pn2 += ys[i] * xs[i]; pd2 += xs[i] * xs[i]; }
    float num2 = block_reduce_sum(pn2, red);
    float den2 = block_reduce_sum(pd2, red);
    float tmp = (num2 - largest * den2) / den2;

    float smallest = tmp + largest;
    float r = largest / smallest - 1.0f;
    if (t == 0) pen[b] = r * r;               // BETA = 1.0
}

// ---------------------------------------------------------------------------
// Kernel 3: deterministic final reduction (no float atomics)
// ---------------------------------------------------------------------------
__global__ __launch_bounds__(64)
void of_finalize(const float* __restrict__ pen, float* __restrict__ out) {
    __shared__ float red[64];
    const int t = threadIdx.x;
    red[t] = pen[t];
    __syncthreads();
    #pragma unroll
    for (int s = 32; s > 0; s >>= 1) {
        if (t < s) red[t] += red[t + s];
        __syncthreads();
    }
    if (t == 0) out[0] = red[0] / (float)B_DIM;
}

extern "C" void kernel_launch(void* const* d_in, const int* in_sizes, int n_in,
                              void* d_out, int out_size, void* d_ws, size_t ws_size,
                              hipStream_t stream) {
    const float* A  = (const float*)d_in[0];   // (64, 512, 28, 28) f32
    const float* x0 = (const float*)d_in[1];   // (64, 512, 1)      f32

    float* AAT = (float*)d_ws;                               // 64*512*512 f32 = 67 MB
    float* pen = AAT + (size_t)B_DIM * C_DIM * C_DIM;        // 64 f32

    dim3 g1(8, 8, B_DIM);                      // mt-groups, ng, batch
    of_aat_wmma<<<g1, 128, 0, stream>>>(A, AAT);
    of_power_iter<<<B_DIM, 256, 0, stream>>>(AAT, x0, pen);
    of_finalize<<<1, 64, 0, stream>>>(pen, (float*)d_out);
}